// BatchKMeans_48996986913304
// MI455X (gfx1250) — compile-verified
//
#include <hip/hip_runtime.h>

// ---------------------------------------------------------------------------
// Batched K-Means for MI455X (gfx1250, wave32).
//   data:      [L][D][N]  f32   (L=8, D=16, N=100000)
//   centroids: [L][D][K]  f32   (K=128)
// Assignment step uses V_WMMA_F32_16X16X4_F32 (full f32 precision; the
// problem is L2-bandwidth bound, ~51MB data resident in 192MB L2, so f32
// WMMA is free and keeps argmax exact).
// ---------------------------------------------------------------------------

typedef float v2f __attribute__((ext_vector_type(2)));
typedef float v8f __attribute__((ext_vector_type(8)));

#define LBATCH 8
#define DDIM   16
#define NPTS   100000
#define KCL    128
#define ITERS  10
#define TILES_PER_L   (NPTS / 16)                 // 6250 (exact)
#define WAVES_PER_BLK 8
#define BLOCKS_X ((TILES_PER_L + WAVES_PER_BLK - 1) / WAVES_PER_BLK)  // 782
#define LDK (LBATCH * DDIM * KCL)                 // 16384

// ---- init: copy input centroids into workspace (don't mutate d_in) --------
__global__ void kmeans_copy_kernel(const float* __restrict__ src,
                                   float* __restrict__ dst, int n) {
  int t = blockIdx.x * blockDim.x + threadIdx.x;
  if (t < n) dst[t] = src[t];
}

// ---- prep: hn[l][k] = -0.5*sum_d c^2 ; zero sums & counts -----------------
__global__ void kmeans_prep_kernel(const float* __restrict__ cents,  // [L][D][K]
                                   float* __restrict__ hn,           // [L][K]
                                   float* __restrict__ gsums,        // [L][D][K]
                                   float* __restrict__ gcounts) {    // [L][K]
  int t = blockIdx.x * blockDim.x + threadIdx.x;
  if (t >= LDK) return;
  gsums[t] = 0.0f;
  int k   = t % KCL;
  int rem = t / KCL;        // l*D + d
  int d   = rem % DDIM;
  int l   = rem / DDIM;
  if (d == 0) {
    const float* c = cents + l * DDIM * KCL + k;
    float s = 0.0f;
#pragma unroll
    for (int dd = 0; dd < DDIM; ++dd) { float x = c[dd * KCL]; s += x * x; }
    hn[l * KCL + k]      = -0.5f * s;
    gcounts[l * KCL + k] = 0.0f;
  }
}

// ---- assign: WMMA-based nearest-centroid + LDS-aggregated scatter ---------
__global__ __launch_bounds__(256)
void kmeans_assign_kernel(const float* __restrict__ data,    // [L][D][N]
                          const float* __restrict__ cents,   // [L][D][K]
                          const float* __restrict__ hn,      // [L][K]
                          int*         __restrict__ labels,  // [L][N]
                          float*       __restrict__ gsums,   // [L][D][K]
                          float*       __restrict__ gcounts) // [L][K]
{
  __shared__ float s_c[DDIM * KCL];     // centroids, [d][k]   (8 KB)
  __shared__ float s_hn[KCL];
  __shared__ float s_sum[KCL * DDIM];   // partial sums, [k][d] (8 KB)
  __shared__ float s_cnt[KCL];

  const int l   = blockIdx.y;
  const int tid = threadIdx.x;

  const float* cl = cents + l * DDIM * KCL;
  for (int i = tid; i < DDIM * KCL; i += 256) { s_c[i] = cl[i]; s_sum[i] = 0.0f; }
  for (int i = tid; i < KCL; i += 256)        { s_hn[i] = hn[l * KCL + i]; s_cnt[i] = 0.0f; }
  __syncthreads();

  const int wave = tid >> 5;
  const int lane = tid & 31;
  const int half = lane >> 4;      // 0: K/M-low half, 1: K/M-high half
  const int col  = lane & 15;      // N = point within tile; also M index for A
  const int tile = blockIdx.x * WAVES_PER_BLK + wave;

  if (tile < TILES_PER_L) {        // wave-uniform -> EXEC all-1s inside
    const int p = tile * 16 + col;
    const float* dl = data + l * DDIM * NPTS;

    // B fragments: B[s] = { data[4s+2*half][p], data[4s+2*half+1][p] }
    v2f B[4];
#pragma unroll
    for (int s = 0; s < 4; ++s) {
      int d0 = 4 * s + 2 * half;
      B[s].x = dl[d0 * NPTS + p];
      B[s].y = dl[(d0 + 1) * NPTS + p];
    }

    float bestv = -3.4e38f;
    int   besti = 0;

#pragma unroll
    for (int ct = 0; ct < KCL / 16; ++ct) {
      const int cbase = ct * 16;
      // C init = -0.5*||c||^2 for cluster cbase + v + 8*half
      v8f acc;
#pragma unroll
      for (int v = 0; v < 8; ++v) acc[v] = s_hn[cbase + v + 8 * half];
      // 4 chained K=4 steps cover d=0..15
#pragma unroll
      for (int s = 0; s < 4; ++s) {
        int d0 = 4 * s + 2 * half;
        v2f A;
        A.x = s_c[d0 * KCL + cbase + col];
        A.y = s_c[(d0 + 1) * KCL + cbase + col];
        acc = __builtin_amdgcn_wmma_f32_16x16x4_f32(
            /*neg_a=*/false, A, /*neg_b=*/false, B[s],
            /*c_mod=*/(short)0, acc, /*reuse_a=*/false, /*reuse_b=*/false);
      }
      // per-lane argmax over the 8 rows this lane holds (ascending index
      // with strict '>' keeps the lowest index on ties, like jnp.argmax)
#pragma unroll
      for (int v = 0; v < 8; ++v) {
        float dv = acc[v];
        if (dv > bestv) { bestv = dv; besti = cbase + v + 8 * half; }
      }
    }

    // combine the two lane-halves (lane p <-> lane p+16 hold the same point)
    float ov = __shfl_xor(bestv, 16, 32);
    int   oi = __shfl_xor(besti, 16, 32);
    if (ov > bestv || (ov == bestv && oi < besti)) { bestv = ov; besti = oi; }

    if (half == 0) labels[l * NPTS + p] = besti;

    // scatter this point's coordinates into the block-local histogram;
    // each lane holds 8 of the 16 dims of its point
#pragma unroll
    for (int s = 0; s < 4; ++s) {
      int d0 = 4 * s + 2 * half;
      atomicAdd(&s_sum[besti * DDIM + d0],     B[s].x);
      atomicAdd(&s_sum[besti * DDIM + d0 + 1], B[s].y);
    }
    if (half == 0) atomicAdd(&s_cnt[besti], 1.0f);
  }
  __syncthreads();

  // flush block-local histogram to global with low-contention atomics
  float* gs = gsums + l * DDIM * KCL;
  for (int i = tid; i < KCL * DDIM; i += 256) {
    int k = i >> 4, d = i & 15;
    float v = s_sum[i];
    if (v != 0.0f) atomicAdd(&gs[d * KCL + k], v);
  }
  for (int i = tid; i < KCL; i += 256) {
    float v = s_cnt[i];
    if (v != 0.0f) atomicAdd(&gcounts[l * KCL + i], v);
  }
}

// ---- update: new centroids = sums / counts --------------------------------
__global__ void kmeans_update_kernel(const float* __restrict__ gsums,
                                     const float* __restrict__ gcounts,
                                     float* __restrict__ cents,
                                     float* __restrict__ out_c,
                                     int write_out) {
  int t = blockIdx.x * blockDim.x + threadIdx.x;
  if (t >= LDK) return;
  int k = t % KCL;
  int l = t / (DDIM * KCL);
  float c = gsums[t] / gcounts[l * KCL + k];   // matches reference (inf/nan on empty)
  cents[t] = c;
  if (write_out) out_c[t] = c;
}

// ---------------------------------------------------------------------------
extern "C" void kernel_launch(void* const* d_in, const int* in_sizes, int n_in,
                              void* d_out, int out_size, void* d_ws, size_t ws_size,
                              hipStream_t stream) {
  const float* data   = (const float*)d_in[0];   // [L][D][N]
  const float* cent0  = (const float*)d_in[1];   // [L][D][K]

  int*   labels_out = (int*)d_out;                        // first 800000 slots
  float* cent_out   = (float*)d_out + LBATCH * NPTS;      // next 16384 slots

  // workspace layout
  float* cbuf   = (float*)d_ws;            // [L][D][K] current centroids
  float* hn     = cbuf + LDK;              // [L][K]    -0.5*||c||^2
  float* gsums  = hn + LBATCH * KCL;       // [L][D][K]
  float* gcnts  = gsums + LDK;             // [L][K]

  kmeans_copy_kernel<<<(LDK + 255) / 256, 256, 0, stream>>>(cent0, cbuf, LDK);

  dim3 agrid(BLOCKS_X, LBATCH);
  for (int it = 0; it < ITERS; ++it) {
    kmeans_prep_kernel<<<(LDK + 255) / 256, 256, 0, stream>>>(cbuf, hn, gsums, gcnts);
    kmeans_assign_kernel<<<agrid, 256, 0, stream>>>(data, cbuf, hn,
                                                    labels_out, gsums, gcnts);
    kmeans_update_kernel<<<(LDK + 255) / 256, 256, 0, stream>>>(
        gsums, gcnts, cbuf, cent_out, it == ITERS - 1 ? 1 : 0);
  }
}